// GCN_81492709475036
// MI455X (gfx1250) — compile-verified
//
#include <hip/hip_runtime.h>
#include <math.h>

// GCN on MI455X (gfx1250, wave32). Matrix ops use V_WMMA_F32_16X16X4_F32.
// Weights staged in LDS (conflict-free transposed layout, ds_load_b64 fragments).
// Aggregation (the HBM/L2-roofline bottleneck, ~10 GB total) is edge-parallel
// float4 gathers + global float atomics.

#define NNODES 50000
#define NEDGES 800000
#define NTILES_M 3125          // 50000 / 16
#define WAVES_PER_BLOCK 8      // 256 threads, wave32
#define BS_STRIDE 260          // 260 % 64 == 4 -> conflict-free b64 fragment reads

typedef float v2f __attribute__((ext_vector_type(2)));
typedef float v8f __attribute__((ext_vector_type(8)));

// ---------------- degree / norm precompute ----------------

__global__ __launch_bounds__(256) void zero_u32_kernel(unsigned int* __restrict__ p, int n) {
    int i = blockIdx.x * blockDim.x + threadIdx.x;
    if (i < n) p[i] = 0u;
}

__global__ __launch_bounds__(256) void count_edges_kernel(const int* __restrict__ col,
                                                          unsigned int* __restrict__ cnt, int e) {
    int i = blockIdx.x * blockDim.x + threadIdx.x;
    if (i < e) atomicAdd(&cnt[col[i]], 1u);
}

// in-place: reinterpret integer degree count -> float dinv = (deg+1)^-1/2
__global__ __launch_bounds__(256) void dinv_kernel(float* __restrict__ p, int n) {
    int i = blockIdx.x * blockDim.x + threadIdx.x;
    if (i < n) {
        unsigned int c = ((const unsigned int*)p)[i];
        p[i] = rsqrtf((float)(c + 1u));   // deg >= 1 always (self loop)
    }
}

__global__ __launch_bounds__(256) void edge_norm_kernel(const int* __restrict__ row,
                                                        const int* __restrict__ col,
                                                        const float* __restrict__ dinv,
                                                        float* __restrict__ norm, int e) {
    int i = blockIdx.x * blockDim.x + threadIdx.x;
    if (i < e) norm[i] = dinv[row[i]] * dinv[col[i]];
}

// ---------------- fp32 WMMA GEMM: C[M,Nc] = act(A[M,K]) @ B[K,Nc] ----------------
// Each wave computes a 16x32 strip (two 16x16 accumulators for ILP).
// The block's K x 32 weight strip is staged once in LDS, transposed
// (Bs[n][k]) with stride 260 so fragment reads are conflict-free ds_load_b64.
// RELU_IN: apply ReLU to A elements on load (fuses previous layer's activation).
// ACT==2:  C = sigmoid(C + bias[n])  (final dense head epilogue).
template<int RELU_IN, int ACT>
__global__ __launch_bounds__(256)
void gemm_wmma_kernel(const float* __restrict__ A, const float* __restrict__ B,
                      float* __restrict__ C, const float* __restrict__ bias,
                      int M, int K, int Nc) {
    __shared__ float Bs[32][BS_STRIDE];                // 33.3 KB of 320 KB WGP LDS

    const int lane = threadIdx.x & 31;
    const int wave = threadIdx.x >> 5;
    const int tileM = (blockIdx.y * WAVES_PER_BLOCK + wave) << 4;
    const int tileN = blockIdx.x << 5;                 // 32-wide strip

    // Cooperative, coalesced load of the full K x 32 weight strip (once).
    for (int idx = threadIdx.x; idx < (K << 5); idx += 256) {
        int n = idx & 31;
        int k = idx >> 5;
        Bs[n][k] = B[(size_t)k * Nc + tileN + n];
    }
    __syncthreads();
    if (tileM >= M) return;                            // wave-uniform (tail waves)

    const int m  = lane & 15;
    const int hs = lane >> 4;                          // half-select (0/1)

    // A fragment (16x4 f32): v0 = A[m][k+2h], v1 = A[m][k+2h+1]  -> float2 load
    const float* arow = A + (size_t)(tileM + m) * K + (hs << 1);
    // B fragment (4x16 f32): v0 = B[k+2h][n], v1 = B[k+2h+1][n]; n = lane&15
    const float* bs0 = &Bs[m][hs << 1];                // acc0: cols tileN + 0..15
    const float* bs1 = &Bs[m + 16][hs << 1];           // acc1: cols tileN + 16..31

    v8f acc0 = {};
    v8f acc1 = {};
#pragma unroll 4
    for (int k = 0; k < K; k += 4) {
        v2f a = *(const v2f*)(arow + k);
        if (RELU_IN) { a.x = fmaxf(a.x, 0.0f); a.y = fmaxf(a.y, 0.0f); }
        v2f b0 = *(const v2f*)(bs0 + k);               // ds_load_b64, conflict-free
        v2f b1 = *(const v2f*)(bs1 + k);
        acc0 = __builtin_amdgcn_wmma_f32_16x16x4_f32(false, a, false, b0, (short)0, acc0, false, false);
        acc1 = __builtin_amdgcn_wmma_f32_16x16x4_f32(false, a, false, b1, (short)0, acc1, false, false);
    }

    // C/D layout: VGPR i, lanes<16 -> M=i, lanes>=16 -> M=i+8; N = lane&15
    const int n0 = tileN + m;
    const int n1 = n0 + 16;
    float bb0 = 0.0f, bb1 = 0.0f;
    if (ACT == 2) { bb0 = bias[n0]; bb1 = bias[n1]; }
#pragma unroll
    for (int i = 0; i < 8; ++i) {
        const int r = tileM + i + (hs << 3);
        float v0 = acc0[i];
        float v1 = acc1[i];
        if (ACT == 2) {
            v0 = 1.0f / (1.0f + expf(-(v0 + bb0)));
            v1 = 1.0f / (1.0f + expf(-(v1 + bb1)));
        }
        C[(size_t)r * Nc + n0] = v0;
        C[(size_t)r * Nc + n1] = v1;
    }
}

// ---------------- aggregation ----------------

// out[i][d] = h[i][d] * dinv[i]^2 + bias[d]   (self-loop term + bias)
__global__ __launch_bounds__(256)
void agg_init_kernel(const float* __restrict__ h, const float* __restrict__ dinv,
                     const float* __restrict__ bias, float* __restrict__ out,
                     unsigned int total, int dshift) {
    unsigned int gid = blockIdx.x * blockDim.x + threadIdx.x;
    if (gid >= total) return;
    unsigned int i = gid >> dshift;
    unsigned int d = gid & ((1u << dshift) - 1u);
    float s = dinv[i];
    out[gid] = h[gid] * (s * s) + bias[d];
}

// per-edge scatter: out[col[e]][4c..4c+3] += h[row[e]][4c..4c+3] * norm[e]
// consecutive threads -> consecutive feature chunks of the same edge (coalesced).
__global__ __launch_bounds__(256)
void agg_edges_kernel(const float* __restrict__ h, const int* __restrict__ row,
                      const int* __restrict__ col, const float* __restrict__ norm,
                      float* __restrict__ out, unsigned int total, int cshift) {
    unsigned int gid = blockIdx.x * blockDim.x + threadIdx.x;
    if (gid >= total) return;
    unsigned int e = gid >> cshift;
    unsigned int c = gid & ((1u << cshift) - 1u);
    float nv = norm[e];
    int r  = row[e];
    int cl = col[e];
    const float4 hv = *((const float4*)(h + ((size_t)r << (cshift + 2))) + c);
    float* op = out + ((size_t)cl << (cshift + 2)) + (c << 2);
    atomicAdd(op + 0, hv.x * nv);
    atomicAdd(op + 1, hv.y * nv);
    atomicAdd(op + 2, hv.z * nv);
    atomicAdd(op + 3, hv.w * nv);
}

// ---------------- host launch ----------------

static inline unsigned int cdiv(unsigned int a, unsigned int b) { return (a + b - 1) / b; }

extern "C" void kernel_launch(void* const* d_in, const int* in_sizes, int n_in,
                              void* d_out, int out_size, void* d_ws, size_t ws_size,
                              hipStream_t stream) {
    const float* x   = (const float*)d_in[0];
    const int*   ei  = (const int*)d_in[1];        // edge_index [2, E], row-major
    const int*   row = ei;
    const int*   col = ei + NEDGES;
    const float* W1 = (const float*)d_in[2]; const float* b1 = (const float*)d_in[3];
    const float* W2 = (const float*)d_in[4]; const float* b2 = (const float*)d_in[5];
    const float* W3 = (const float*)d_in[6]; const float* b3 = (const float*)d_in[7];
    const float* W4 = (const float*)d_in[8]; const float* b4 = (const float*)d_in[9];
    float* out = (float*)d_out;

    // workspace layout (floats), all offsets 16B-aligned:
    float* ws   = (float*)d_ws;
    float* dinv = ws;                                   // N
    float* norm = ws + NNODES;                          // E
    float* buf0 = norm + NEDGES;                        // N*256
    float* buf1 = buf0 + (size_t)NNODES * 256;          // N*256

    const int T = 256;
    const unsigned int M = NNODES, E = NEDGES;

    // 1) degree (exact integer count) -> dinv -> per-edge norm
    zero_u32_kernel<<<cdiv(M, T), T, 0, stream>>>((unsigned int*)dinv, M);
    count_edges_kernel<<<cdiv(E, T), T, 0, stream>>>(col, (unsigned int*)dinv, E);
    dinv_kernel<<<cdiv(M, T), T, 0, stream>>>(dinv, M);
    edge_norm_kernel<<<cdiv(E, T), T, 0, stream>>>(row, col, dinv, norm, E);

    const dim3 blk(T);
    const unsigned int gy = cdiv(NTILES_M, WAVES_PER_BLOCK);   // 391

    // One GCN layer: tmp = act(in) @ W ; out = scatter(tmp) + selfloop + bias.
    // 'in' is dead after the GEMM, so out may alias in (never aliases tmp).
    auto gcn_layer = [&](const float* in, const float* W, const float* b,
                         float* tmp, float* o, int K, int D, bool relu_in) {
        dim3 grid(D / 32, gy);
        if (relu_in)
            gemm_wmma_kernel<1, 0><<<grid, blk, 0, stream>>>(in, W, tmp, nullptr, M, K, D);
        else
            gemm_wmma_kernel<0, 0><<<grid, blk, 0, stream>>>(in, W, tmp, nullptr, M, K, D);
        int dshift = (D == 256) ? 8 : 6;
        unsigned int tot = M << dshift;
        agg_init_kernel<<<cdiv(tot, T), blk, 0, stream>>>(tmp, dinv, b, o, tot, dshift);
        int cshift = (D == 256) ? 6 : 4;                 // chunks = D/4
        unsigned int te = E << cshift;
        agg_edges_kernel<<<cdiv(te, T), blk, 0, stream>>>(tmp, row, col, norm, o, te, cshift);
    };

    // layer 1: x -> buf1   (ReLU deferred into next layer's GEMM)
    gcn_layer(x, W1, b1, buf0, buf1, 256, 256, false);
    // layer 2 applied 4x with shared weights
    for (int i = 0; i < 4; ++i)
        gcn_layer(buf1, W2, b2, buf0, buf1, 256, 256, true);
    // layer 3: 256 -> 64, no output ReLU
    gcn_layer(buf1, W3, b3, buf0, buf1, 256, 64, true);
    // dense head: sigmoid(h @ W4 + b4) -> d_out
    dim3 g4(64 / 32, gy);
    gemm_wmma_kernel<0, 2><<<g4, blk, 0, stream>>>(buf1, W4, out, b4, M, 64, 64);
}